// Model_35072702939636
// MI455X (gfx1250) — compile-verified
//
#include <hip/hip_runtime.h>
#include <hip/hip_bf16.h>

typedef float v2f __attribute__((ext_vector_type(2)));
typedef float v8f __attribute__((ext_vector_type(8)));

#define H 4096
#define W 4096

// Pass 1: zero the output frame with 128-bit stores (64 MB of float4s).
__global__ __launch_bounds__(256) void zero_out_kernel(float4* __restrict__ out) {
    size_t idx = (size_t)blockIdx.x * blockDim.x + threadIdx.x;
    out[idx] = make_float4(0.f, 0.f, 0.f, 0.f);
}

// Pass 2: one wave32 per 16x16 pixel tile. Three V_WMMA_F32_16X16X4_F32 ops
// evaluate numx/numy/den for all 256 pixels of the tile; each lane then does
// IEEE divisions, trunc, clamp, and 8 scatter stores.
__global__ __launch_bounds__(256) void warp_scatter_kernel(
    const float* __restrict__ X, const float* __restrict__ wts,
    float* __restrict__ out)
{
    const float p0 = wts[0], p1 = wts[1], p2 = wts[2], p3 = wts[3];
    const float p4 = wts[4], p5 = wts[5], p6 = wts[6], p7 = wts[7];

    const int lane = threadIdx.x & 31;
    const int wv   = threadIdx.x >> 5;
    const int tile = blockIdx.x * 8 + wv;          // 65536 tiles total
    const int i0   = (tile >> 8) << 4;             // tile row origin
    const int j0   = (tile & 255) << 4;            // tile col origin
    const int half = lane >> 4;                    // 0: lanes 0-15, 1: lanes 16-31
    const int n    = lane & 15;                    // column within tile
    const float fj = (float)(j0 + n);

    // A (16x4 f32): lanes 0-15 hold K=0,1 -> [i, 1]; lanes 16-31 hold K=2,3 -> 0.
    // B (4x16 f32): lanes 0-15 hold K=0,1 -> [coef_i, c0 + coef_j*j]; lanes 16-31 -> 0.
    // D[m][n] = coef_i*(i0+m) + c0 + coef_j*(j0+n)  — the full affine form.
    const float m0 = (half == 0) ? 1.0f : 0.0f;    // branch-free masking, EXEC stays full
    v2f A, Bx, By, Bd;
    A.x  = m0 * (float)(i0 + n);  A.y  = m0;
    Bx.x = m0 * p2;               Bx.y = m0 * (p0 + p3 * fj);
    By.x = m0 * p4;               By.y = m0 * (p1 + p5 * fj);
    Bd.x = m0 * p6;               Bd.y = m0 * (1.0f + p7 * fj);

    v8f z = {};
    v8f numx = __builtin_amdgcn_wmma_f32_16x16x4_f32(false, A, false, Bx, (short)0, z, false, false);
    v8f numy = __builtin_amdgcn_wmma_f32_16x16x4_f32(false, A, false, By, (short)0, z, false, false);
    v8f den  = __builtin_amdgcn_wmma_f32_16x16x4_f32(false, A, false, Bd, (short)0, z, false, false);

    // D layout: lane L holds column N=L%16, rows M = v + 8*(L/16) in VGPR v.
    // Load the matching source pixels: per load, lanes 0-15 read 64 contiguous
    // bytes of row i0+v, lanes 16-31 read 64 contiguous bytes of row i0+8+v.
    const float* src = X + ((size_t)(i0 + half * 8)) * W + (size_t)(j0 + n);
    float xv[8];
#pragma unroll
    for (int v = 0; v < 8; ++v) xv[v] = src[(size_t)v * W];

#pragma unroll
    for (int v = 0; v < 8; ++v) {
        const float d  = den[v];
        float qx = truncf(numx[v] / d);            // IEEE div keeps trunc exact
        float qy = truncf(numy[v] / d);
        qx = fminf(fmaxf(qx, 0.0f), (float)(H - 1));
        qy = fminf(fmaxf(qy, 0.0f), (float)(W - 1));
        const int xi = (int)qx;
        const int yi = (int)qy;
        out[(size_t)xi * W + yi] = xv[v];          // scatter; L2 (192MB) absorbs it
    }
}

extern "C" void kernel_launch(void* const* d_in, const int* in_sizes, int n_in,
                              void* d_out, int out_size, void* d_ws, size_t ws_size,
                              hipStream_t stream) {
    const float* X   = (const float*)d_in[0];
    const float* wts = (const float*)d_in[1];
    float* out = (float*)d_out;

    // 4096*4096 floats / 4 per thread / 256 per block = 16384 blocks
    zero_out_kernel<<<(H * W / 4) / 256, 256, 0, stream>>>((float4*)out);

    // 256*256 tiles / 8 tiles per block = 8192 blocks of 8 waves
    warp_scatter_kernel<<<(H / 16) * (W / 16) / 8, 256, 0, stream>>>(X, wts, out);
}